// LPC_19284403159085
// MI455X (gfx1250) — compile-verified
//
#include <hip/hip_runtime.h>
#include <hip/hip_bf16.h>
#include <math.h>

typedef float v2f __attribute__((ext_vector_type(2)));
typedef float v8f __attribute__((ext_vector_type(8)));

#define BATCH 32
#define TLEN  480000
#define NF    3001      // 1 + T/HOP
#define HOP   160
#define FFTN  512
#define NBINS 257
#define NFILT 21
#define NPAD  272       // 17 tiles of 16 bins
#define ORD   13
#define FSTR  516       // LDS row stride for frames (bank-spread)

// ---- ws float offsets ----
#define WS_Z      0                         // 32*21*3001 = 2016672
#define WS_WTS    2016672                   // 21*272 = 5712
#define WS_COS    (WS_WTS + 5712)           // 512
#define WS_WIN    (WS_COS + 512)            // 512
#define WS_EQL    (WS_WIN + 512)            // 32
#define WS_RC     (WS_EQL + 32)             // 14*19 -> 272

// ---------------------------------------------------------------------------
// Kernel 0: constant tables
// ---------------------------------------------------------------------------
__global__ void tables_kernel(float* __restrict__ wts, float* __restrict__ costab,
                              float* __restrict__ win, float* __restrict__ eql033,
                              float* __restrict__ rcoef) {
  const int tid = threadIdx.x;              // 256 threads
  const float PI = 3.14159265358979323846f;
  // cos table + padded Hann window (WINLEN=400, centered in 512)
  for (int i = tid; i < FFTN; i += 256) {
    costab[i] = cosf(2.0f * PI * (float)i / (float)FFTN);
    float wv = 0.0f;
    if (i >= 56 && i < 456) {
      float nn = (float)(i - 56);
      wv = 0.5f * (1.0f - cosf(2.0f * PI * nn / 400.0f));
    }
    win[i] = wv;
  }
  const float nyqbark = 6.0f * asinhf(8000.0f / 600.0f);
  const float step = nyqbark / 20.0f;       // nfilts-1 = 20
  // Bark filterbank weights, wts[band][k], padded to 272 cols
  for (int i = tid; i < NFILT * NPAD; i += 256) {
    int band = i / NPAD, k = i % NPAD;
    float v = 0.0f;
    if (k < NBINS) {
      float bb  = 6.0f * asinhf((31.25f * (float)k) / 600.0f);  // SR/NFFT = 31.25
      float mid = (float)band * step;
      float lof = bb - mid - 0.5f;
      float hif = bb - mid + 0.5f;
      float e = fminf(0.0f, fminf(hif, -2.5f * lof));
      v = exp2f(e * 3.3219280948873623f);   // 10^e
    }
    wts[i] = v;
  }
  // equal loudness ^0.33
  if (tid < NFILT) {
    float cf  = 600.0f * sinhf((float)tid * step / 6.0f);
    float fsq = cf * cf;
    float e = fsq / (fsq + 160000.0f);
    e = e * e * (fsq + 1440000.0f) / (fsq + 9610000.0f);
    eql033[tid] = powf(e, 0.33f);
  }
  // autocorr coefficients: r[k] = sum_{j=1..19} rcoef[k][j-1]*z[j]
  // (40-pt symmetric IDFT + edge-band duplication folded in)
  for (int i = tid; i < 14 * 19; i += 256) {
    int k = i / 19, j = i % 19 + 1;
    float c = 2.0f * cosf(2.0f * PI * (float)(j * k) / 40.0f);
    if (j == 1)  c += 1.0f;                       // x[0]   = z[1]
    if (j == 19) c += (k & 1) ? -1.0f : 1.0f;     // x[20]  = z[19], cos(pi*k)
    rcoef[k * 19 + (j - 1)] = c / 40.0f;
  }
}

// ---------------------------------------------------------------------------
// Kernel 1: windowed DFT power spectrum via f32 WMMA + Bark + log
// One block = 16 frames of one batch. 4 waves, each owns N-tiles of 16 bins.
// ---------------------------------------------------------------------------
__global__ __launch_bounds__(128) void dft_bark_kernel(
    const float* __restrict__ wave, float* __restrict__ Z,
    const float* __restrict__ wts, const float* __restrict__ costab,
    const float* __restrict__ win) {
  __shared__ float sF[16 * FSTR];   // windowed frames 16 x 512 (stride 516)
  __shared__ float sT[FFTN];        // cos table
  __shared__ float sP[16 * NPAD];   // power spectrum 16 x 272

  const int tid   = threadIdx.x;
  const int b     = blockIdx.y;
  const int fbase = blockIdx.x * 16;
  const float* w  = wave + (size_t)b * TLEN;

  for (int i = tid; i < FFTN; i += 128) sT[i] = costab[i];

  // stage frames (reflect padding, window applied)
  for (int i = tid; i < 16 * FFTN; i += 128) {
    int m = i >> 9, k = i & 511;
    long s = (long)(fbase + m) * HOP + k - 256;
    if (s < 0) s = -s;
    if (s >= TLEN) s = 2L * TLEN - 2 - s;
    sF[m * FSTR + k] = w[s] * win[k];
  }
  __syncthreads();

  const int wv   = tid >> 5;
  const int lane = tid & 31;
  const int ln   = lane & 15;
  const int half = lane >> 4;       // 0: K+0,1 ; 1: K+2,3 (A and B row halves)
  const int o    = half * 2;

  for (int t = wv; t < 17; t += 4) {
    v8f cre = {}; v8f cim = {};
    const int n = t * 16 + ln;      // bin column
    for (int kk = 0; kk < 128; kk++) {
      const int k0 = kk * 4 + o;
      v2f a;  a.x = sF[ln * FSTR + k0];     a.y = sF[ln * FSTR + k0 + 1];
      const int i1 = (k0 * n) & 511;
      const int i2 = ((k0 + 1) * n) & 511;
      v2f bc; bc.x = sT[i1];                bc.y = sT[i2];
      v2f bs; bs.x = sT[(i1 + 384) & 511];  bs.y = sT[(i2 + 384) & 511];
      cre = __builtin_amdgcn_wmma_f32_16x16x4_f32(false, a, false, bc,
                                                  (short)0, cre, false, false);
      cim = __builtin_amdgcn_wmma_f32_16x16x4_f32(false, a, false, bs,
                                                  (short)0, cim, false, false);
    }
    // C layout: vgpr r, lanes0-15 -> frame r, lanes16-31 -> frame r+8
    #pragma unroll
    for (int r = 0; r < 8; r++) {
      float re = cre[r], im = cim[r];
      sP[(r + half * 8) * NPAD + n] = re * re + im * im + 400.0f;  // + WINLEN
    }
  }
  __syncthreads();

  // Bark matvec + natural log -> Z[b][band][f]
  for (int oi = tid; oi < 16 * NFILT; oi += 128) {
    int frame = oi / NFILT, band = oi % NFILT;
    int gf = fbase + frame;
    if (gf >= NF) continue;
    const float* wr = wts + band * NPAD;
    float acc = 0.0f;
    for (int k = 0; k < NBINS; k++) acc += wr[k] * sP[frame * NPAD + k];
    Z[((size_t)b * NFILT + band) * NF + gf] = logf(acc);
  }
}

// ---------------------------------------------------------------------------
// Kernel 2: RASTA filter (in-place on Z), then z = exp(0.33*y)*eql^0.33
// ---------------------------------------------------------------------------
__global__ void rasta_kernel(float* __restrict__ Z, const float* __restrict__ eql033) {
  int g = blockIdx.x * 128 + threadIdx.x;
  if (g >= BATCH * NFILT) return;
  const int band = g % NFILT;
  const float e33 = eql033[band];
  const size_t base = (size_t)g * NF;
  float x0 = Z[base + 0], x1 = Z[base + 1], x2 = Z[base + 2], x3 = Z[base + 3];
  Z[base + 0] = e33; Z[base + 1] = e33; Z[base + 2] = e33; Z[base + 3] = e33;
  float y = 0.0f;
  for (int n = 4; n < NF; n++) {
    float xn = Z[base + n];
    float fv = 0.2f * xn + 0.1f * x3 - 0.1f * x1 - 0.2f * x0;
    y = fv + 0.94f * y;
    Z[base + n] = expf(0.33f * y) * e33;
    x0 = x1; x1 = x2; x2 = x3; x3 = xn;
  }
}

// ---------------------------------------------------------------------------
// Kernel 3: autocorrelation lags + order-13 Levinson-Durbin per (b, frame)
// ---------------------------------------------------------------------------
__global__ __launch_bounds__(128) void lev_kernel(const float* __restrict__ Z,
                                                  const float* __restrict__ rcoef,
                                                  float* __restrict__ out) {
  __shared__ float sR[14 * 19];
  const int tid = threadIdx.x;
  for (int i = tid; i < 14 * 19; i += 128) sR[i] = rcoef[i];
  __syncthreads();
  const int f = blockIdx.x * 128 + tid;
  const int b = blockIdx.y;
  if (f >= NF) return;

  float zv[19];
  #pragma unroll
  for (int j = 0; j < 19; j++)
    zv[j] = Z[((size_t)b * NFILT + (j + 1)) * NF + f];

  float r[14];
  #pragma unroll
  for (int k = 0; k < 14; k++) {
    float a = 0.0f;
    #pragma unroll
    for (int j = 0; j < 19; j++) a += sR[k * 19 + j] * zv[j];
    r[k] = a;
  }

  float P = r[0];
  float A[13];
  #pragma unroll
  for (int k = 0; k < 13; k++) {
    float acc = r[k + 1];
    #pragma unroll
    for (int i = 0; i < 13; i++) if (i < k) acc += A[i] * r[k - i];
    float t = -acc / P;
    P *= (1.0f - t * t);
    float Ao[13];
    #pragma unroll
    for (int i = 0; i < 13; i++) if (i < k) Ao[i] = A[i];
    #pragma unroll
    for (int i = 0; i < 13; i++) if (i < k) A[i] = Ao[i] + t * Ao[k - 1 - i];
    A[k] = t;
  }
  const float inv = 1.0f / (P + 1e-8f);
  const size_t ob = (size_t)b * ORD * NF + f;
  out[ob] = inv;
  #pragma unroll
  for (int c = 1; c < ORD; c++) out[ob + (size_t)c * NF] = A[c - 1] * inv;
}

// ---------------------------------------------------------------------------
extern "C" void kernel_launch(void* const* d_in, const int* in_sizes, int n_in,
                              void* d_out, int out_size, void* d_ws, size_t ws_size,
                              hipStream_t stream) {
  const float* wave = (const float*)d_in[0];
  float* out = (float*)d_out;
  float* ws  = (float*)d_ws;

  float* Zb     = ws + WS_Z;
  float* wts    = ws + WS_WTS;
  float* costab = ws + WS_COS;
  float* win    = ws + WS_WIN;
  float* eql    = ws + WS_EQL;
  float* rcoef  = ws + WS_RC;

  tables_kernel<<<1, 256, 0, stream>>>(wts, costab, win, eql, rcoef);
  dft_bark_kernel<<<dim3((NF + 15) / 16, BATCH), 128, 0, stream>>>(wave, Zb, wts, costab, win);
  rasta_kernel<<<(BATCH * NFILT + 127) / 128, 128, 0, stream>>>(Zb, eql);
  lev_kernel<<<dim3((NF + 127) / 128, BATCH), 128, 0, stream>>>(Zb, rcoef, out);
}